// GATConv_85693187490236
// MI455X (gfx1250) — compile-verified
//
#include <hip/hip_runtime.h>
#include <cstdint>

#define N_NODES 8192
#define IN_FEATS 256
#define OUT_FEATS 128
#define ALPHA 0.2f
#define ADJ_WORDS 256            // N_NODES / 32 bitmask words per row
#define P_STRIDE 68              // 16x64 P tile, padded stride to avoid LDS bank conflicts
#define MAX_COLS 8192            // worst-case union column list

typedef __attribute__((ext_vector_type(2))) float v2f;
typedef __attribute__((ext_vector_type(8))) float v8f;

__device__ __forceinline__ v8f wmma_f32x4(v2f a, v2f b, v8f c) {
  // D = A(16x4 f32) * B(4x16 f32) + C(16x16 f32), wave32
  return __builtin_amdgcn_wmma_f32_16x16x4_f32(false, a, false, b, (short)0, c,
                                               false, false);
}

// ---------------------------------------------------------------------------
// Kernel 1: zero adjacency bitmask (ws is not re-poisoned between replays)
// ---------------------------------------------------------------------------
__global__ void gat_zero_adj(unsigned* __restrict__ adj, int nwords) {
  int i = blockIdx.x * blockDim.x + threadIdx.x;
  if (i < nwords) adj[i] = 0u;
}

// ---------------------------------------------------------------------------
// Kernel 2: scatter edge list into bitmask (atomicOr is idempotent => dedup
// of duplicate random edges, matching the dense to_dense_adj semantics)
// ---------------------------------------------------------------------------
__global__ void gat_scatter(const int* __restrict__ ei, unsigned* __restrict__ adj,
                            int E) {
  int e = blockIdx.x * blockDim.x + threadIdx.x;
  if (e >= E) return;
  int i = ei[e];       // edge_index[0, e]  (source row of the mask)
  int j = ei[E + e];   // edge_index[1, e]
  atomicOr(&adj[(size_t)i * ADJ_WORDS + (j >> 5)], 1u << (j & 31));
}

// ---------------------------------------------------------------------------
// Kernel 3: Wh = h @ W via fp32 WMMA, one wave per 16x16 output tile
// ---------------------------------------------------------------------------
__global__ __launch_bounds__(32) void gat_gemm_wh(const float* __restrict__ h,
                                                  const float* __restrict__ W,
                                                  float* __restrict__ Wh) {
  const int lane  = threadIdx.x;
  const int mtile = blockIdx.x >> 3;   // 512 row tiles
  const int ntile = blockIdx.x & 7;    // 8 col tiles
  const int m  = lane & 15;
  const int kh = (lane >> 4) << 1;     // lanes 0-15 -> K {0,1}; 16-31 -> K {2,3}
  const float* hrow = h + (size_t)(mtile * 16 + m) * IN_FEATS;
  const int ncol = ntile * 16 + m;

  v8f acc = {0.f, 0.f, 0.f, 0.f, 0.f, 0.f, 0.f, 0.f};
  for (int k = 0; k < IN_FEATS; k += 4) {
    v2f a;
    a.x = hrow[k + kh];
    a.y = hrow[k + kh + 1];
    v2f b;
    b.x = W[(size_t)(k + kh) * OUT_FEATS + ncol];
    b.y = W[(size_t)(k + kh + 1) * OUT_FEATS + ncol];
    acc = wmma_f32x4(a, b, acc);
  }
  const int rbase = mtile * 16 + ((lane >> 4) << 3);
#pragma unroll
  for (int r = 0; r < 8; ++r)
    Wh[(size_t)(rbase + r) * OUT_FEATS + ntile * 16 + m] = acc[r];
}

// ---------------------------------------------------------------------------
// Kernel 4: Wh1 = Wh @ a[:128], Wh2 = Wh @ a[128:]
// ---------------------------------------------------------------------------
__global__ void gat_scores(const float* __restrict__ Wh, const float* __restrict__ av,
                           float* __restrict__ wh1, float* __restrict__ wh2) {
  int i = blockIdx.x * blockDim.x + threadIdx.x;
  if (i >= N_NODES) return;
  const float* r = Wh + (size_t)i * OUT_FEATS;
  float s1 = 0.f, s2 = 0.f;
  for (int k = 0; k < OUT_FEATS; ++k) {
    float v = r[k];
    s1 += v * av[k];
    s2 += v * av[OUT_FEATS + k];
  }
  wh1[i] = s1;
  wh2[i] = s2;
}

// ---------------------------------------------------------------------------
// Kernel 5: fused masked softmax + sparse-gathered P @ Wh
// One 256-thread block (8 waves) per 16-row tile; each wave owns 16 output cols.
// ---------------------------------------------------------------------------
__global__ __launch_bounds__(256) void gat_attn_fused(
    const float* __restrict__ Wh, const float* __restrict__ wh1,
    const float* __restrict__ wh2, const unsigned* __restrict__ adj,
    float* __restrict__ out) {
  __shared__ unsigned s_adj[16 * ADJ_WORDS];        // 16 KB: 16 rows of bitmask
  __shared__ unsigned short s_cols[MAX_COLS];       // 16 KB: union column list
  __shared__ float s_P[16 * P_STRIDE];              // 4.25 KB: gathered P tile
  __shared__ float s_wh1[16];
  __shared__ float s_m[16];
  __shared__ float s_linv[16];
  __shared__ float s_redm[256];
  __shared__ float s_redl[256];
  __shared__ int s_off[256];
  __shared__ unsigned s_union[256];
  __shared__ int s_nc, s_ncp;

  const int t  = threadIdx.x;
  const int r0 = blockIdx.x * 16;

  // ---- load tile state -----------------------------------------------------
  for (int idx = t; idx < 16 * ADJ_WORDS; idx += 256)
    s_adj[idx] = adj[(size_t)r0 * ADJ_WORDS + idx];
  if (t < 16) s_wh1[t] = wh1[r0 + t];
  __syncthreads();

  // ---- pass 1: per-row online softmax stats over set bits only ------------
  const int row = t & 15;
  const int grp = t >> 4;    // each row split across 16 threads x 16 words
  {
    float wm = -INFINITY, wl = 0.f;
    const float b1 = s_wh1[row];
    for (int w = grp * 16; w < grp * 16 + 16; ++w) {
      unsigned word = s_adj[row * ADJ_WORDS + w];
      while (word) {
        int b = __builtin_ctz(word);
        word &= word - 1;
        float v = b1 + wh2[(w << 5) + b];
        v = v > 0.f ? v : ALPHA * v;
        if (v > wm) { wl = wl * __expf(wm - v) + 1.f; wm = v; }
        else        { wl += __expf(v - wm); }
      }
    }
    s_redm[row * 16 + grp] = wm;
    s_redl[row * 16 + grp] = wl;
  }
  // union of the 16 rows' bitmask, one word per thread
  {
    unsigned u = 0u;
#pragma unroll
    for (int r = 0; r < 16; ++r) u |= s_adj[r * ADJ_WORDS + t];
    s_union[t] = u;
  }
  __syncthreads();

  if (t < 16) {  // combine 16 partial (m,l) pairs per row
    float M = -INFINITY, L = 0.f;
    for (int g = 0; g < 16; ++g) {
      float mg = s_redm[t * 16 + g], lg = s_redl[t * 16 + g];
      if (lg > 0.f) {
        if (mg > M) { L = L * __expf(M - mg) + lg; M = mg; }
        else        { L += lg * __expf(mg - M); }
      }
    }
    s_m[t]    = M;
    s_linv[t] = 1.f / L;   // every row has a self-loop => L > 0
  }
  if (t == 0) {  // serial prefix scan of popcounts (256 adds, negligible)
    int acc = 0;
    for (int w = 0; w < 256; ++w) {
      s_off[w] = acc;
      acc += __popc(s_union[w]);
    }
    s_nc  = acc;
    s_ncp = (acc + 63) & ~63;   // pad K to WMMA-chunk multiple
  }
  __syncthreads();

  // ---- emit union column list (sparse gather index) ------------------------
  {
    int off = s_off[t];
    unsigned word = s_union[t];
    while (word) {
      int b = __builtin_ctz(word);
      word &= word - 1;
      s_cols[off++] = (unsigned short)((t << 5) + b);
    }
  }
  const int nc  = s_nc;
  const int ncp = s_ncp;
  for (int idx = nc + t; idx < ncp; idx += 256) s_cols[idx] = 0;  // safe pad
  __syncthreads();

  // ---- pass 2: out_tile = P(16 x nc) @ Wh[cols](nc x 128) via fp32 WMMA ---
  const int lane = t & 31;
  const int wave = t >> 5;
  const int am   = lane & 15;          // A row / B col within tile
  const int ak   = (lane >> 4) << 1;   // K sub-offset {0,2}
  const int ncol = wave * 16 + am;
  v8f acc = {0.f, 0.f, 0.f, 0.f, 0.f, 0.f, 0.f, 0.f};

  for (int kc = 0; kc < ncp; kc += 64) {
    // fill P tile: 16 rows x 64 gathered columns, 4 values per thread
#pragma unroll
    for (int u = 0; u < 4; ++u) {
      int idx = (t << 2) + u;
      int pr  = idx & 15;
      int pc  = idx >> 4;
      int gk  = kc + pc;
      float p = 0.f;
      if (gk < nc) {
        int col = s_cols[gk];
        if ((s_adj[pr * ADJ_WORDS + (col >> 5)] >> (col & 31)) & 1u) {
          float v = s_wh1[pr] + wh2[col];
          v = v > 0.f ? v : ALPHA * v;
          p = __expf(v - s_m[pr]) * s_linv[pr];
        }
      }
      s_P[pr * P_STRIDE + pc] = p;
    }
    __syncthreads();

#pragma unroll 4
    for (int ks = 0; ks < 16; ++ks) {
      int kb = (ks << 2) + ak;
      v2f a;
      a.x = s_P[am * P_STRIDE + kb];
      a.y = s_P[am * P_STRIDE + kb + 1];
      v2f b;  // gathered rows of Wh (padding rows multiply a zero P => 0)
      b.x = Wh[(size_t)s_cols[kc + kb] * OUT_FEATS + ncol];
      b.y = Wh[(size_t)s_cols[kc + kb + 1] * OUT_FEATS + ncol];
      acc = wmma_f32x4(a, b, acc);
    }
    __syncthreads();
  }

  const int rbase = r0 + ((lane >> 4) << 3);
#pragma unroll
  for (int r = 0; r < 8; ++r)
    out[(size_t)(rbase + r) * OUT_FEATS + ncol] = acc[r];
}

// ---------------------------------------------------------------------------
extern "C" void kernel_launch(void* const* d_in, const int* in_sizes, int n_in,
                              void* d_out, int out_size, void* d_ws, size_t ws_size,
                              hipStream_t stream) {
  const float* h  = (const float*)d_in[0];
  const int*   ei = (const int*)d_in[1];
  const float* W  = (const float*)d_in[2];
  const float* av = (const float*)d_in[3];
  float* out = (float*)d_out;
  const int E = in_sizes[1] / 2;

  // workspace layout: Wh (4 MB) | wh1 (32 KB) | wh2 (32 KB) | adj bitmask (8 MB)
  float* Wh  = (float*)d_ws;
  float* wh1 = Wh + (size_t)N_NODES * OUT_FEATS;
  float* wh2 = wh1 + N_NODES;
  unsigned* adj = (unsigned*)(wh2 + N_NODES);
  const int nwords = N_NODES * ADJ_WORDS;

  gat_zero_adj<<<(nwords + 255) / 256, 256, 0, stream>>>(adj, nwords);
  gat_scatter<<<(E + 255) / 256, 256, 0, stream>>>(ei, adj, E);
  gat_gemm_wh<<<(N_NODES / 16) * (OUT_FEATS / 16), 32, 0, stream>>>(h, W, Wh);
  gat_scores<<<N_NODES / 256, 256, 0, stream>>>(Wh, av, wh1, wh2);
  gat_attn_fused<<<N_NODES / 16, 256, 0, stream>>>(Wh, wh1, wh2, adj, out);
}